// HeavySnow_42906723287182
// MI455X (gfx1250) — compile-verified
//
#include <hip/hip_runtime.h>
#include <cstdint>

#define IMH 1024
#define IMW 1024
#define NFLAKES 15728
#define NB 16
#define NC 3
#define LST 20  // LDS stride of 20x20 input tile (floats)
#define VST 20  // LDS stride of 16x20 intermediate tile (floats)

typedef __attribute__((ext_vector_type(2))) float v2f;
typedef __attribute__((ext_vector_type(8))) float v8f;
typedef __attribute__((address_space(1))) int as1_int;
typedef __attribute__((address_space(3))) int as3_int;

#if defined(__has_builtin)
#if __has_builtin(__builtin_amdgcn_global_load_async_to_lds_b32)
#define HAVE_ASYNC_LDS 1
#endif
#endif

__device__ __forceinline__ float gval(int t, float G0, float G1, float G2) {
  float r = 0.0f;
  r = (t == 0) ? G0 : r;
  r = (t == 1) ? G1 : r;
  r = (t == 2) ? G2 : r;
  r = (t == 3) ? G1 : r;
  r = (t == 4) ? G0 : r;
  return r;
}

__device__ __forceinline__ v8f wmma4(v2f a, v2f b, v8f c) {
  // D = A(16x4) * B(4x16) + C ;  8 args: neg_a, A, neg_b, B, c_mod, C, reuse_a, reuse_b
  return __builtin_amdgcn_wmma_f32_16x16x4_f32(false, a, false, b, (short)0, c,
                                               false, false);
}

// ---------------- kernel 1: x -> ws (float4 copy) ----------------
__global__ __launch_bounds__(256) void HeavySnow_copy(const float4* __restrict__ s,
                                                      float4* __restrict__ d) {
  const size_t i = (size_t)blockIdx.x * 256 + threadIdx.x;
  d[i] = s[i];
}

// ---------------- kernel 2: stamp flakes into ws ----------------
// thread = (flake, one of 49 offsets). Same-value races are benign (all 0.95).
__global__ __launch_bounds__(256) void HeavySnow_stamp(const int* __restrict__ ys,
                                                       const int* __restrict__ xs,
                                                       const int* __restrict__ rs,
                                                       float* __restrict__ ws) {
  const int tid = blockIdx.x * 256 + threadIdx.x;
  if (tid >= NB * NFLAKES * 49) return;
  const int f = tid / 49;
  const int o = tid - f * 49;
  const int q = o / 7;
  const int dy = q - 3;
  const int dx = (o - q * 7) - 3;
  const int r = rs[f];
  if (dy > r || dy < -r || dx > r || dx < -r) return;  // invalid -> center already stamped
  const int b = f / NFLAKES;
  int yy = ys[f] + dy;
  yy = yy < 0 ? 0 : (yy > IMH - 1 ? IMH - 1 : yy);
  int xx = xs[f] + dx;
  xx = xx < 0 ? 0 : (xx > IMW - 1 ? IMW - 1 : xx);
  const size_t base = (size_t)b * NC * IMH * IMW + (size_t)yy * IMW + xx;
  ws[base] = 0.95f;
  ws[base + (size_t)IMH * IMW] = 0.95f;
  ws[base + 2 * (size_t)IMH * IMW] = 0.95f;
}

// ---------------- kernel 3: 5x5 gaussian blur via WMMA + clip ----------------
// One 16x16 output tile per wave; 4 waves (128 threads) per block.
__global__ __launch_bounds__(128) void HeavySnow_blur(const float* __restrict__ src,
                                                      float* __restrict__ dst) {
  __shared__ float sIn[4][20 * LST];  // 20x20 haloed input tile per wave
  __shared__ float sV[4][16 * VST];   // 16x20 vertically blurred tile per wave

  const int lane = threadIdx.x & 31;
  const int wv = threadIdx.x >> 5;
  const int t = blockIdx.x * 4 + wv;  // 196608 tiles, grid covers exactly
  const int tx = t & 63;
  const int ty = (t >> 6) & 63;
  const int pl = t >> 12;  // plane = b*3 + c, 0..47
  const float* __restrict__ plane = src + (size_t)pl * (IMH * IMW);
  float* __restrict__ outp = dst + (size_t)pl * (IMH * IMW);
  const int C0 = tx << 4;
  const int R0 = ty << 4;
  float* In = sIn[wv];
  float* V = sV[wv];

  // Gaussian weights, sigma=1.5, K=5 (computed like the reference, then normalized)
  const float w0 = expf(-4.0f / 4.5f);
  const float w1 = expf(-1.0f / 4.5f);
  const float sw = 1.0f + 2.0f * w1 + 2.0f * w0;
  const float G0 = w0 / sw, G1 = w1 / sw, G2 = 1.0f / sw;

  // Per-lane weight operand, shared by both passes:
  //  pass V (A-side): A[m][k0+k] = Wv[m][kr] = g[kr-m]   (m = lane&15)
  //  pass H (B-side): B[k][n]    = Wh[kr][n] = g[kr-n]   (n = lane&15)
  const int mn = lane & 15;
  const int h2 = (lane >> 4) << 1;
  v2f wk[5];
#pragma unroll
  for (int cI = 0; cI < 5; ++cI) {
    const int kr = 4 * cI + h2;
    wk[cI].x = gval(kr - mn, G0, G1, G2);
    wk[cI].y = gval(kr + 1 - mn, G0, G1, G2);
  }

  // ---- stage 20x20 input tile (zero-padded halo) into LDS ----
  const bool interior = (tx > 0) && (tx < 63) && (ty > 0) && (ty < 63);
#if defined(HAVE_ASYNC_LDS)
  if (interior) {
    const float* gbase = plane + (size_t)(R0 - 2) * IMW + (C0 - 2);
#pragma unroll
    for (int rnd = 0; rnd < 13; ++rnd) {
      const int idx = rnd * 32 + lane;
      if (idx < 400) {
        const int r = idx / 20;
        const int c = idx - r * 20;
        __builtin_amdgcn_global_load_async_to_lds_b32(
            (as1_int*)(gbase + r * IMW + c),
            (as3_int*)(&In[r * LST + c]), 0, 0);
      }
    }
#if __has_builtin(__builtin_amdgcn_s_wait_asynccnt)
    __builtin_amdgcn_s_wait_asynccnt(0);
#else
    asm volatile("s_wait_asynccnt 0" ::: "memory");
#endif
  } else
#endif
  {
#pragma unroll
    for (int rnd = 0; rnd < 13; ++rnd) {
      const int idx = rnd * 32 + lane;
      if (idx < 400) {
        const int r = idx / 20;
        const int c = idx - r * 20;
        const int gr = R0 - 2 + r;
        const int gc = C0 - 2 + c;
        float v = 0.0f;
        if (gr >= 0 && gr < IMH && gc >= 0 && gc < IMW) v = plane[(size_t)gr * IMW + gc];
        In[r * LST + c] = v;
      }
    }
  }
  __syncthreads();

  // ---- pass 1 (vertical): V(16x20) = Wv(16x20) x In(20x20), two N-halves ----
  v8f accL = {0.f, 0.f, 0.f, 0.f, 0.f, 0.f, 0.f, 0.f};
  v8f accR = {0.f, 0.f, 0.f, 0.f, 0.f, 0.f, 0.f, 0.f};
  {
    const int col = lane & 15;
#pragma unroll
    for (int cI = 0; cI < 5; ++cI) {
      const int kr = 4 * cI + h2;  // B rows: lanes0-15 -> k0,k0+1 ; lanes16-31 -> k0+2,k0+3
      v2f bL, bR;
      bL.x = In[kr * LST + col];
      bL.y = In[(kr + 1) * LST + col];
      bR.x = In[kr * LST + col + 4];
      bR.y = In[(kr + 1) * LST + col + 4];
      accL = wmma4(wk[cI], bL, accL);
      accR = wmma4(wk[cI], bR, accR);
    }
  }
  // scatter V (D-layout: VGPR v -> row v + 8*(lane>=16), col = lane&15) to LDS
  {
    const int col = lane & 15;
    const int rbase = (lane >> 4) << 3;
#pragma unroll
    for (int v = 0; v < 8; ++v) V[(rbase + v) * VST + col] = accL[v];
    if (col >= 12) {  // accR covers cols 4..19; only 16..19 are new
#pragma unroll
      for (int v = 0; v < 8; ++v) V[(rbase + v) * VST + 4 + col] = accR[v];
    }
  }
  __syncthreads();

  // ---- pass 2 (horizontal): Out(16x16) = V(16x20) x Wh(20x16) ----
  v8f acc = {0.f, 0.f, 0.f, 0.f, 0.f, 0.f, 0.f, 0.f};
  {
#pragma unroll
    for (int cI = 0; cI < 5; ++cI) {
      const int kr = 4 * cI + h2;                      // A cols: k0+2*(lane>=16) + {0,1}
      v2f a = *(const v2f*)&V[mn * VST + kr];          // 8B-aligned contiguous pair
      acc = wmma4(a, wk[cI], acc);
    }
  }
  // clip + store (D-layout)
  {
    const int col = C0 + (lane & 15);
    const int rbase = R0 + ((lane >> 4) << 3);
#pragma unroll
    for (int v = 0; v < 8; ++v) {
      const float val = fminf(fmaxf(acc[v], 0.0f), 1.0f);
      outp[(size_t)(rbase + v) * IMW + col] = val;
    }
  }
}

extern "C" void kernel_launch(void* const* d_in, const int* in_sizes, int n_in,
                              void* d_out, int out_size, void* d_ws, size_t ws_size,
                              hipStream_t stream) {
  const float* x = (const float*)d_in[0];
  const int* ys = (const int*)d_in[1];
  const int* xs = (const int*)d_in[2];
  const int* rs = (const int*)d_in[3];
  float* ws = (float*)d_ws;  // stamped image, NB*NC*IMH*IMW floats
  float* out = (float*)d_out;

  // 1) copy x -> ws : 50331648 floats = 12582912 float4 = 49152 blocks x 256
  HeavySnow_copy<<<49152, 256, 0, stream>>>((const float4*)x, (float4*)ws);
  // 2) stamp: 16*15728*49 = 12330752 threads = 48167 blocks x 256 (exact)
  HeavySnow_stamp<<<48167, 256, 0, stream>>>(ys, xs, rs, ws);
  // 3) blur: 16*3*64*64 = 196608 tiles, 4 tiles/block -> 49152 blocks x 128
  HeavySnow_blur<<<49152, 128, 0, stream>>>(ws, out);
}